// TransformerEdgeClassifier_89721866813770
// MI455X (gfx1250) — compile-verified
//
#include <hip/hip_runtime.h>
#include <math.h>

// ---------------------------------------------------------------------------
// TransformerConv GNN edge classifier for gfx1250 (MI455X).
// wave32 + V_WMMA_F32_16X16X4_F32 GEMMs + TDM (tensor_load_to_lds) weight
// staging + L2-resident float atomics for the segment softmax/scatter.
// ---------------------------------------------------------------------------

typedef __attribute__((ext_vector_type(2))) float    v2f;
typedef __attribute__((ext_vector_type(8))) float    v8f;
typedef __attribute__((ext_vector_type(4))) unsigned uv4;
typedef __attribute__((ext_vector_type(8))) int      iv8;
typedef __attribute__((ext_vector_type(4))) int      iv4;

#define NN   50000
#define EE   800000
#define EMBD 128         // din for every layer (and K of all node GEMMs)
#define CH   32          // per-head channels
#define TPB  256         // 8 wave32s per block
#define MT   3125        // NN/16 row tiles (exact)

#if __has_builtin(__builtin_amdgcn_tensor_load_to_lds) && \
    __has_builtin(__builtin_amdgcn_s_wait_tensorcnt)
#define USE_TDM 1
#endif

// ----------------------------- utility fills --------------------------------
__global__ void fill_f32(float* __restrict__ p, float v, int n) {
    int i = blockIdx.x * blockDim.x + threadIdx.x;
    if (i < n) p[i] = v;
}
__global__ void fill_u32(unsigned* __restrict__ p, unsigned v, int n) {
    int i = blockIdx.x * blockDim.x + threadIdx.x;
    if (i < n) p[i] = v;
}

// ------------------------------ embedding -----------------------------------
__global__ void embed_k(const int* __restrict__ x, const float* __restrict__ emb,
                        float* __restrict__ h) {
    int i = blockIdx.x * blockDim.x + threadIdx.x;
    if (i >= NN * EMBD) return;
    int node = i >> 7, c = i & 127;
    int code = x[node * 2];                  // x[:,0]
    h[i] = emb[code * EMBD + c];
}

// ------------------- fp32 WMMA GEMM: C = A(Mx128) * W(128xN) + b ------------
// Grid: (ceil(MT/8), Ndim/16). Each block stages the 128x16 weight panel for
// its tn column into LDS once (TDM async DMA), then 8 waves each compute one
// 16x16 tile with V_WMMA_F32_16X16X4_F32, B served from LDS.
__global__ void gemm_bias_wmma(const float* __restrict__ A, const float* __restrict__ W,
                               const float* __restrict__ bias, float* __restrict__ C,
                               int Ndim) {
    __shared__ float shW[EMBD * 16];         // [k][n] panel, 8 KB
    int wid = threadIdx.x >> 5, lane = threadIdx.x & 31;
    int tn = blockIdx.y;

#if defined(USE_TDM)
    if (wid == 0) {
        unsigned lds_off = (unsigned)(size_t)(&shW[0]);          // flat low 32 = LDS offset
        unsigned long long ga = (unsigned long long)(size_t)(W + tn * 16);
        uv4 g0 = {};
        g0[0] = 1u;                                              // count=1, user mode
        g0[1] = lds_off;                                         // lds_addr
        g0[2] = (unsigned)(ga & 0xffffffffu);                    // global_addr[31:0]
        g0[3] = (unsigned)((ga >> 32) & 0x1ffffffu) | (2u << 30);// addr[56:32] | type=2
        iv8 g1 = {};
        g1[0] = (int)(2u << 16);                                 // data_size = 4 bytes
        g1[1] = (int)(((unsigned)Ndim & 0xffffu) << 16);         // tensor_dim0 lo
        g1[2] = (int)((((unsigned)Ndim >> 16) & 0xffffu) |
                      (((unsigned)EMBD & 0xffffu) << 16));       // dim0 hi | dim1 lo
        g1[3] = (int)(((unsigned)EMBD >> 16) & 0xffffu) | (16 << 16); // dim1 hi | tile_dim0=16
        g1[4] = (int)(EMBD & 0xffff);                            // tile_dim1=128, tile_dim2=0
        g1[5] = (int)Ndim;                                       // tensor_dim0_stride lo
        g1[6] = 0;
        g1[7] = 0;
        iv4 z4 = {};
        iv8 z8 = {};
        __builtin_amdgcn_tensor_load_to_lds(g0, g1, z4, z4, z8, 0);
        __builtin_amdgcn_s_wait_tensorcnt(0);
    }
#else
    // Fallback: cooperative staged load of the 128x16 panel.
    for (int i = threadIdx.x; i < EMBD * 16; i += TPB) {
        int k = i >> 4, n = i & 15;
        shW[i] = W[(size_t)k * Ndim + tn * 16 + n];
    }
#endif
    __syncthreads();

    int tm = blockIdx.x * (TPB / 32) + wid;  // uniform per wave
    if (tm >= MT) return;
    int ml = lane & 15, half = lane >> 4;

    const float* Arow = A + (size_t)(tm * 16 + ml) * EMBD;
    v8f acc = {};
#pragma unroll
    for (int kk = 0; kk < EMBD; kk += 4) {
        int ka = kk + half * 2;
        v2f a, b;
        a.x = Arow[ka];
        a.y = Arow[ka + 1];
        b.x = shW[ka * 16 + ml];
        b.y = shW[(ka + 1) * 16 + ml];
        acc = __builtin_amdgcn_wmma_f32_16x16x4_f32(false, a, false, b,
                                                    (short)0, acc, false, false);
    }
    float bc = bias[tn * 16 + ml];
#pragma unroll
    for (int v = 0; v < 8; ++v) {
        int row = tm * 16 + v + 8 * half;
        C[(size_t)row * Ndim + tn * 16 + ml] = acc[v] + bc;
    }
}

// -------------------- monotonic float <-> u32 (for atomicMax) ---------------
__device__ __forceinline__ unsigned f32_enc(float f) {
    unsigned b = __float_as_uint(f);
    return (b & 0x80000000u) ? ~b : (b | 0x80000000u);
}
__device__ __forceinline__ float f32_dec(unsigned u) {
    return (u & 0x80000000u) ? __uint_as_float(u ^ 0x80000000u)
                             : __uint_as_float(~u);
}

// ------------------- attention: alpha + segment max -------------------------
__global__ void attn_alpha(const int* __restrict__ src, const int* __restrict__ dst,
                           const float* __restrict__ ea,
                           const float* __restrict__ Q, const float* __restrict__ K,
                           const float* __restrict__ We,
                           float* __restrict__ alpha, unsigned* __restrict__ amaxU,
                           int H, int dout, float inv_sqrt_c, int total) {
    int idx = blockIdx.x * blockDim.x + threadIdx.x;
    if (idx >= total) return;
    int e = idx / H, hh = idx - e * H;
    int s = src[e], d = dst[e];
    float a0 = ea[3 * e], a1 = ea[3 * e + 1], a2 = ea[3 * e + 2];
    const float* qr = Q + (size_t)d * dout + hh * CH;
    const float* kr = K + (size_t)s * dout + hh * CH;
    const float* w0 = We + hh * CH;
    const float* w1 = w0 + dout;
    const float* w2 = w1 + dout;
    float acc = 0.f;
#pragma unroll
    for (int c = 0; c < CH; ++c) {
        float eec = fmaf(a0, w0[c], fmaf(a1, w1[c], a2 * w2[c]));
        acc = fmaf(qr[c], kr[c] + eec, acc);
    }
    acc *= inv_sqrt_c;
    alpha[idx] = acc;
    atomicMax(&amaxU[d * H + hh], f32_enc(acc));
}

// ------------------- attention: exp + segment sum ---------------------------
__global__ void attn_expsum(const int* __restrict__ dst, float* __restrict__ alpha,
                            const unsigned* __restrict__ amaxU,
                            float* __restrict__ denom, int H, int total) {
    int idx = blockIdx.x * blockDim.x + threadIdx.x;
    if (idx >= total) return;
    int e = idx / H, hh = idx - e * H;
    int d = dst[e];
    float m = f32_dec(amaxU[d * H + hh]);
    float ex = __expf(alpha[idx] - m);
    alpha[idx] = ex;
    atomicAdd(&denom[d * H + hh], ex);
}

// ---------- attention: weighted message scatter (float4 per thread) ---------
__global__ void attn_scatter(const int* __restrict__ src, const int* __restrict__ dst,
                             const float* __restrict__ ea, const float* __restrict__ V,
                             const float* __restrict__ We, const float* __restrict__ alpha,
                             const float* __restrict__ denom, float* __restrict__ acc,
                             int H, int dout, int total4) {
    int idx = blockIdx.x * blockDim.x + threadIdx.x;
    if (idx >= total4) return;
    int per = dout >> 2;                      // quads per edge
    int e = idx / per, q = idx - e * per;
    int col = q * 4;
    int hh = col >> 5;                        // quad never straddles a head
    int s = src[e], d = dst[e];
    float w = alpha[e * H + hh] / fmaxf(denom[d * H + hh], 1e-16f);
    float a0 = ea[3 * e], a1 = ea[3 * e + 1], a2 = ea[3 * e + 2];
    const float4 v4 = *(const float4*)(V + (size_t)s * dout + col);
    const float4 e0 = *(const float4*)(We + col);
    const float4 e1 = *(const float4*)(We + dout + col);
    const float4 e2 = *(const float4*)(We + 2 * dout + col);
    float* base = acc + (size_t)d * dout + col;
    atomicAdd(base + 0, (v4.x + fmaf(a0, e0.x, fmaf(a1, e1.x, a2 * e2.x))) * w);
    atomicAdd(base + 1, (v4.y + fmaf(a0, e0.y, fmaf(a1, e1.y, a2 * e2.y))) * w);
    atomicAdd(base + 2, (v4.z + fmaf(a0, e0.z, fmaf(a1, e1.z, a2 * e2.z))) * w);
    atomicAdd(base + 3, (v4.w + fmaf(a0, e0.w, fmaf(a1, e1.w, a2 * e2.w))) * w);
}

// ---------- combine + LayerNorm + ReLU + optional residual (in-place h) -----
__global__ void combine_ln(const float* __restrict__ acc, const float* __restrict__ S,
                           const float* __restrict__ g, const float* __restrict__ bb,
                           float* __restrict__ h, int dnorm, int residual, int nrows) {
    int wid = threadIdx.x >> 5, lane = threadIdx.x & 31;
    int row = blockIdx.x * (blockDim.x >> 5) + wid;
    if (row >= nrows) return;
    int per = dnorm >> 5;                     // 4 (d=128) or 1 (d=32)
    float t[4];
    float sum = 0.f;
    for (int i = 0; i < per; ++i) {
        int c = i * 32 + lane;
        t[i] = acc[(size_t)row * dnorm + c] + S[(size_t)row * dnorm + c];
        sum += t[i];
    }
    for (int off = 16; off; off >>= 1) sum += __shfl_xor(sum, off, 32);
    float mu = sum / (float)dnorm;
    float vs = 0.f;
    for (int i = 0; i < per; ++i) { float dlt = t[i] - mu; vs += dlt * dlt; }
    for (int off = 16; off; off >>= 1) vs += __shfl_xor(vs, off, 32);
    float inv = rsqrtf(vs / (float)dnorm + 1e-5f);
    for (int i = 0; i < per; ++i) {
        int c = i * 32 + lane;
        float y = fmaf((t[i] - mu) * inv, g[c], bb[c]);
        y = fmaxf(y, 0.f);
        if (residual) y += h[(size_t)row * dnorm + c];
        h[(size_t)row * dnorm + c] = y;
    }
}

// -------- final edge MLP: concat(h[src],h[dst]) @ W1 -> ReLU -> @ W2 --------
__global__ void edge_mlp(const int* __restrict__ src, const int* __restrict__ dst,
                         const float* __restrict__ h,
                         const float* __restrict__ W1, const float* __restrict__ b1,
                         const float* __restrict__ W2, const float* __restrict__ b2,
                         float* __restrict__ out, int ntiles) {
    __shared__ float zsh[TPB / 32][16][33];
    int wid = threadIdx.x >> 5, lane = threadIdx.x & 31;
    int tile = blockIdx.x * (blockDim.x >> 5) + wid;
    if (tile >= ntiles) return;               // grids divide exactly; never taken
    int ml = lane & 15, half = lane >> 4;
    int e = tile * 16 + ml;
    const float* hs = h + (size_t)src[e] * CH;
    const float* hd = h + (size_t)dst[e] * CH;

    v8f acc0 = {}, acc1 = {};
#pragma unroll
    for (int kk = 0; kk < 64; kk += 4) {
        int ka = kk + half * 2;               // even -> no 32-boundary straddle
        v2f a, bA, bB;
        a.x = (ka     < 32) ? hs[ka]     : hd[ka - 32];
        a.y = (ka + 1 < 32) ? hs[ka + 1] : hd[ka - 31];
        bA.x = W1[ka * 32 + ml];       bA.y = W1[(ka + 1) * 32 + ml];
        bB.x = W1[ka * 32 + 16 + ml];  bB.y = W1[(ka + 1) * 32 + 16 + ml];
        acc0 = __builtin_amdgcn_wmma_f32_16x16x4_f32(false, a, false, bA,
                                                     (short)0, acc0, false, false);
        acc1 = __builtin_amdgcn_wmma_f32_16x16x4_f32(false, a, false, bB,
                                                     (short)0, acc1, false, false);
    }
#pragma unroll
    for (int v = 0; v < 8; ++v) {
        int r = v + 8 * half;
        zsh[wid][r][ml]      = fmaxf(acc0[v] + b1[ml],      0.f);
        zsh[wid][r][16 + ml] = fmaxf(acc1[v] + b1[16 + ml], 0.f);
    }
    __syncthreads();
    if (lane < 16) {
        int e2 = tile * 16 + lane;
        float o0 = b2[0], o1 = b2[1], o2 = b2[2];
#pragma unroll
        for (int k2 = 0; k2 < 32; ++k2) {
            float zv = zsh[wid][lane][k2];
            o0 = fmaf(zv, W2[k2 * 3 + 0], o0);
            o1 = fmaf(zv, W2[k2 * 3 + 1], o1);
            o2 = fmaf(zv, W2[k2 * 3 + 2], o2);
        }
        out[(size_t)e2 * 3 + 0] = o0;
        out[(size_t)e2 * 3 + 1] = o1;
        out[(size_t)e2 * 3 + 2] = o2;
    }
}

// ---------------------------------------------------------------------------
static inline int cdiv(int a, int b) { return (a + b - 1) / b; }

extern "C" void kernel_launch(void* const* d_in, const int* in_sizes, int n_in,
                              void* d_out, int out_size, void* d_ws, size_t ws_size,
                              hipStream_t stream) {
    (void)in_sizes; (void)n_in; (void)out_size; (void)ws_size;

    // Inputs: x, edge_index, edge_attr, then params leaves in JAX pytree
    // (sorted-key) order: W1, W2, b1, b2, embed, layers[0..2] each as
    // {We,Wk,Wq,Ws,Wv,bk,bq,bs,bv,ln_b,ln_g}.
    const int*   xin  = (const int*)d_in[0];
    const int*   eidx = (const int*)d_in[1];
    const int*   srcI = eidx;
    const int*   dstI = eidx + EE;
    const float* ea   = (const float*)d_in[2];
    const float* W1   = (const float*)d_in[3];
    const float* W2   = (const float*)d_in[4];
    const float* b1   = (const float*)d_in[5];
    const float* b2   = (const float*)d_in[6];
    const float* embW = (const float*)d_in[7];

    // Workspace (~168 MB of f32)
    float*    ws    = (float*)d_ws;
    float*    h     = ws;                            // N x 128 (shrinks to N x 32)
    float*    acc   = h   + (size_t)NN * EMBD;       // N x dout accumulator
    float*    Q     = acc + (size_t)NN * EMBD;
    float*    Kf    = Q   + (size_t)NN * EMBD;
    float*    V     = Kf  + (size_t)NN * EMBD;
    float*    S     = V   + (size_t)NN * EMBD;       // skip branch
    float*    alpha = S   + (size_t)NN * EMBD;       // E x H
    unsigned* amaxU = (unsigned*)(alpha + (size_t)EE * 4);  // N x H
    float*    denom = (float*)(amaxU + (size_t)NN * 4);     // N x H

    dim3 blk(TPB);
    embed_k<<<cdiv(NN * EMBD, TPB), blk, 0, stream>>>(xin, embW, h);

    const int   Hs[3]    = {4, 4, 1};
    const int   douts[3] = {128, 128, 32};
    const int   resid[3] = {1, 1, 0};
    const float inv_sqrt_c = 0.17677669529663687f;   // 1/sqrt(32)

    for (int L = 0; L < 3; ++L) {
        int base = 8 + L * 11;
        const float* We  = (const float*)d_in[base + 0];
        const float* Wk  = (const float*)d_in[base + 1];
        const float* Wq  = (const float*)d_in[base + 2];
        const float* Wsk = (const float*)d_in[base + 3];
        const float* Wv  = (const float*)d_in[base + 4];
        const float* bk  = (const float*)d_in[base + 5];
        const float* bq  = (const float*)d_in[base + 6];
        const float* bs  = (const float*)d_in[base + 7];
        const float* bv  = (const float*)d_in[base + 8];
        const float* lnb = (const float*)d_in[base + 9];
        const float* lng = (const float*)d_in[base + 10];
        int H = Hs[L], dout = douts[L];

        fill_f32<<<cdiv(NN * dout, TPB), blk, 0, stream>>>(acc,   0.f, NN * dout);
        fill_f32<<<cdiv(NN * H,    TPB), blk, 0, stream>>>(denom, 0.f, NN * H);
        fill_u32<<<cdiv(NN * H,    TPB), blk, 0, stream>>>(amaxU, 0u,  NN * H);

        dim3 ggrid(cdiv(MT, TPB / 32), dout / 16);
        gemm_bias_wmma<<<ggrid, blk, 0, stream>>>(h, Wq,  bq, Q,  dout);
        gemm_bias_wmma<<<ggrid, blk, 0, stream>>>(h, Wk,  bk, Kf, dout);
        gemm_bias_wmma<<<ggrid, blk, 0, stream>>>(h, Wv,  bv, V,  dout);
        gemm_bias_wmma<<<ggrid, blk, 0, stream>>>(h, Wsk, bs, S,  dout);

        attn_alpha<<<cdiv(EE * H, TPB), blk, 0, stream>>>(
            srcI, dstI, ea, Q, Kf, We, alpha, amaxU, H, dout, inv_sqrt_c, EE * H);
        attn_expsum<<<cdiv(EE * H, TPB), blk, 0, stream>>>(
            dstI, alpha, amaxU, denom, H, EE * H);
        attn_scatter<<<cdiv(EE * (dout / 4), TPB), blk, 0, stream>>>(
            srcI, dstI, ea, V, We, alpha, denom, acc, H, dout, EE * (dout / 4));

        combine_ln<<<cdiv(NN, TPB / 32), blk, 0, stream>>>(
            acc, S, lng, lnb, h, dout, resid[L], NN);
    }

    edge_mlp<<<cdiv(EE / 16, TPB / 32), blk, 0, stream>>>(
        srcI, dstI, h, W1, b1, W2, b2, (float*)d_out, EE / 16);
}